// ModelName_14783277432874
// MI455X (gfx1250) — compile-verified
//
#include <hip/hip_runtime.h>
#include <math.h>

#define NODES 100000
#define NEDGE 1600000
#define FDIM  128
#define HDIM  64
#define ODIM  32
#define NEG_SLOPE 0.2f

typedef float v2f __attribute__((ext_vector_type(2)));
typedef float v8f __attribute__((ext_vector_type(8)));
typedef unsigned int v4u __attribute__((ext_vector_type(4)));
typedef int v8i __attribute__((ext_vector_type(8)));
typedef int v4i __attribute__((ext_vector_type(4)));

// ---------------------------------------------------------------------------
// C[M x (NTILES*16)] = A[M x K] @ B[K x (NTILES*16)] (+ bias)
// - B (<=32KB) is staged into LDS once per block by the Tensor Data Mover
//   (tensor_load_to_lds + s_wait_tensorcnt), then read via ds loads.
// - One wave per 16-row stripe; A fragment reused across all NTILES col tiles.
// - V_WMMA_F32_16X16X4_F32 (fp32 matrix pipe, exact accumulation).
//
// A 16x4 f32 layout (ISA 7.12.2): lanes 0-15 hold M=lane with K={0,1} in
// v0/v1; lanes 16-31 hold M=lane-16 with K={2,3}. B mirrors with N=lane%16.
// C/D: VGPR i -> M = i + 8*(lane>=16), N = lane%16.
// ---------------------------------------------------------------------------
template <int NTILES>
__global__ __launch_bounds__(256) void gat_gemm_wmma_f32(
        const float* __restrict__ A,
        const float* __restrict__ B,
        float* __restrict__ C,
        const float* __restrict__ bias,
        int M, int K) {
    const int Nc  = NTILES * 16;
    const int tid = (int)threadIdx.x;

    __shared__ float Bs[FDIM * HDIM];   // 32 KB; only shared object -> LDS offset 0

#if defined(__AMDGCN__) && __has_builtin(__builtin_amdgcn_tensor_load_to_lds)
    if (tid < 32) {
        // TDM D# for a 2D tile copy: tile[K x Nc] f32, row-major, into LDS@0.
        const unsigned long long gaddr = (unsigned long long)B;
        v4u g0;
        g0.x = 1u;                                            // count=1, user mode
        g0.y = 0u;                                            // lds_addr = 0
        g0.z = (unsigned)(gaddr & 0xFFFFFFFFull);             // global_addr[31:0]
        g0.w = (unsigned)((gaddr >> 32) & 0x01FFFFFFull)      // global_addr[56:32]
               | (2u << 30);                                  // type = 2 (image)
        v8i g1;
        g1[0] = (int)(2u << 16);                              // data_size = 4 bytes
        g1[1] = (int)(((unsigned)Nc & 0xFFFFu) << 16);        // tensor_dim0[15:0]
        g1[2] = (int)(((unsigned)K  & 0xFFFFu) << 16);        // tensor_dim1[15:0]
        g1[3] = (int)((unsigned)Nc << 16);                    // tile_dim0 = Nc
        g1[4] = K;                                            // tile_dim1 = K
        g1[5] = Nc;                                           // tensor_dim0_stride
        g1[6] = 0;
        g1[7] = 0;
        v4i g2 = {0, 0, 0, 0};                                // 2D tensor: unused
        v4i g3 = {0, 0, 0, 0};                                // 2D tensor: unused
        v8i g4 = {0, 0, 0, 0, 0, 0, 0, 0};                    // clang-23 extra group
        __builtin_amdgcn_tensor_load_to_lds(g0, g1, g2, g3, g4, 0);
        __builtin_amdgcn_s_wait_tensorcnt(0);
    }
#else
    for (int i = tid; i < K * Nc; i += 256) Bs[i] = B[i];
#endif
    __syncthreads();

    const int wave = (int)blockIdx.x * 8 + (tid >> 5);
    const int lane = tid & 31;
    if (wave * 16 < M) {
        const int row   = wave * 16 + (lane & 15);
        const int khalf = (lane >> 4) << 1;   // 0 or 2
        const int colin = lane & 15;

        v8f acc[NTILES] = {};

        const float* __restrict__ arow = A + (size_t)row * (size_t)K;
        const float* __restrict__ brow = Bs + khalf * Nc + colin;
        for (int k = 0; k < K; k += 4) {
            v2f a;
            a.x = arow[k + khalf];
            a.y = arow[k + khalf + 1];
            const float* bk = brow + k * Nc;
#pragma unroll
            for (int t = 0; t < NTILES; ++t) {
                v2f b;
                b.x = bk[t * 16];
                b.y = bk[Nc + t * 16];
                acc[t] = __builtin_amdgcn_wmma_f32_16x16x4_f32(
                    false, a, false, b, (short)0, acc[t], false, false);
            }
        }

        const int rbase = wave * 16 + ((lane >> 4) << 3);  // +8 for upper lanes
#pragma unroll
        for (int t = 0; t < NTILES; ++t) {
            const int col  = t * 16 + colin;
            const float bb = bias ? bias[col] : 0.0f;
#pragma unroll
            for (int i = 0; i < 8; ++i) {
                C[(size_t)(rbase + i) * Nc + col] = acc[t][i] + bb;
            }
        }
    }
}

// ---------------------------------------------------------------------------
// Per-node attention scores: a_src[n] = dot(h[n], att_src), same for dst.
// One wave per node, H=64 -> 2 floats per lane, butterfly reduce.
// ---------------------------------------------------------------------------
__global__ void gat_scores(const float* __restrict__ h,
                           const float* __restrict__ att_src,
                           const float* __restrict__ att_dst,
                           float* __restrict__ asrc,
                           float* __restrict__ adst) {
    const int wave = (int)((blockIdx.x * blockDim.x + threadIdx.x) >> 5);
    const int lane = (int)(threadIdx.x & 31);
    if (wave >= NODES) return;
    const float* __restrict__ hp = h + (size_t)wave * HDIM;
    const float h0 = hp[lane], h1 = hp[lane + 32];
    float ss = h0 * att_src[lane] + h1 * att_src[lane + 32];
    float dd = h0 * att_dst[lane] + h1 * att_dst[lane + 32];
#pragma unroll
    for (int off = 16; off > 0; off >>= 1) {
        ss += __shfl_xor(ss, off, 32);
        dd += __shfl_xor(dd, off, 32);
    }
    if (lane == 0) { asrc[wave] = ss; adst[wave] = dd; }
}

__global__ void gat_init_nodes(float* __restrict__ m, float* __restrict__ s) {
    const int i = (int)(blockIdx.x * blockDim.x + threadIdx.x);
    if (i < NODES) { m[i] = -INFINITY; s[i] = 0.0f; }
}

__global__ void gat_init_agg(float* __restrict__ agg, const float* __restrict__ bias) {
    const int i = (int)(blockIdx.x * blockDim.x + threadIdx.x);
    if (i < NODES * HDIM) agg[i] = bias[i & (HDIM - 1)];
}

// e[j] = leaky_relu(a_src[src]+a_dst[dst]); segment-max into m[dst]
// (float atomic max via signed/unsigned int monotonic trick; m init -inf)
__global__ void gat_edge_max(const int* __restrict__ src,
                             const int* __restrict__ dst,
                             const float* __restrict__ asrc,
                             const float* __restrict__ adst,
                             float* __restrict__ e,
                             float* __restrict__ m) {
    const int j = (int)(blockIdx.x * blockDim.x + threadIdx.x);
    if (j >= NEDGE) return;
    const int sj = src[j], dj = dst[j];
    float v = asrc[sj] + adst[dj];
    v = (v > 0.0f) ? v : NEG_SLOPE * v;
    e[j] = v;
    if (v >= 0.0f) atomicMax((int*)(m + dj), __float_as_int(v));
    else           atomicMin((unsigned int*)(m + dj), __float_as_uint(v));
}

// e[j] <- exp(e[j] - m[dst]); segment-sum into s[dst]
__global__ void gat_edge_exp(const int* __restrict__ dst,
                             float* __restrict__ e,
                             const float* __restrict__ m,
                             float* __restrict__ s) {
    const int j = (int)(blockIdx.x * blockDim.x + threadIdx.x);
    if (j >= NEDGE) return;
    const int dj = dst[j];
    const float ex = expf(e[j] - m[dj]);
    e[j] = ex;
    atomicAdd(s + dj, ex);
}

// agg[dst] += h[src] * (e[j] / (s[dst]+eps)); 16 lanes per edge, float4 each
__global__ void gat_edge_agg(const int* __restrict__ src,
                             const int* __restrict__ dst,
                             const float* __restrict__ e,
                             const float* __restrict__ s,
                             const float* __restrict__ h,
                             float* __restrict__ agg) {
    const long long gid = (long long)blockIdx.x * blockDim.x + threadIdx.x;
    const int j = (int)(gid >> 4);
    if (j >= NEDGE) return;
    const int f  = ((int)gid & 15) << 2;
    const int sj = src[j], dj = dst[j];
    const float alpha = e[j] / (s[dj] + 1e-16f);
    const float4 hv = *(const float4*)(h + (size_t)sj * HDIM + f);
    float* ap = agg + (size_t)dj * HDIM + f;
    atomicAdd(ap + 0, hv.x * alpha);
    atomicAdd(ap + 1, hv.y * alpha);
    atomicAdd(ap + 2, hv.z * alpha);
    atomicAdd(ap + 3, hv.w * alpha);
}

__global__ void gat_relu(float* __restrict__ x, int n) {
    const int i = (int)(blockIdx.x * blockDim.x + threadIdx.x);
    if (i < n) x[i] = fmaxf(x[i], 0.0f);
}

// log_softmax over 32 classes: one wave per node, one class per lane
__global__ void gat_logsoftmax(const float* __restrict__ logits,
                               float* __restrict__ out) {
    const int wave = (int)((blockIdx.x * blockDim.x + threadIdx.x) >> 5);
    const int lane = (int)(threadIdx.x & 31);
    if (wave >= NODES) return;
    const float v = logits[(size_t)wave * ODIM + lane];
    float mx = v;
#pragma unroll
    for (int off = 16; off > 0; off >>= 1) mx = fmaxf(mx, __shfl_xor(mx, off, 32));
    float sum = expf(v - mx);
#pragma unroll
    for (int off = 16; off > 0; off >>= 1) sum += __shfl_xor(sum, off, 32);
    out[(size_t)wave * ODIM + lane] = v - mx - logf(sum);
}

// ---------------------------------------------------------------------------
extern "C" void kernel_launch(void* const* d_in, const int* in_sizes, int n_in,
                              void* d_out, int out_size, void* d_ws, size_t ws_size,
                              hipStream_t stream) {
    (void)in_sizes; (void)n_in; (void)out_size; (void)ws_size;

    const float* x        = (const float*)d_in[0];
    const int*   ei1      = (const int*)d_in[1];   // [2,E]: src row then dst row
    const int*   ei2      = (const int*)d_in[2];
    const float* W1       = (const float*)d_in[3];
    const float* att_src1 = (const float*)d_in[4];
    const float* att_dst1 = (const float*)d_in[5];
    const float* b1       = (const float*)d_in[6];
    const float* W2       = (const float*)d_in[7];
    const float* att_src2 = (const float*)d_in[8];
    const float* att_dst2 = (const float*)d_in[9];
    const float* b2       = (const float*)d_in[10];
    const float* Wlin     = (const float*)d_in[11];
    const float* blin     = (const float*)d_in[12];
    float*       out      = (float*)d_out;

    // Workspace carve (floats): 2*N*H + 4*N + E + N*ODIM  (~72 MB)
    float* hbuf   = (float*)d_ws;                        // N*H (h1, then h2)
    float* aggbuf = hbuf   + (size_t)NODES * HDIM;       // N*H (agg1/x2, agg2)
    float* asrc   = aggbuf + (size_t)NODES * HDIM;       // N
    float* adst   = asrc   + NODES;                      // N
    float* mbuf   = adst   + NODES;                      // N
    float* sbuf   = mbuf   + NODES;                      // N
    float* ebuf   = sbuf   + NODES;                      // E
    float* logits = ebuf   + NEDGE;                      // N*ODIM

    const dim3 blk(256);
    const int waves_rows   = NODES / 16;                       // 6250 (exact)
    const int gemm_blocks  = (waves_rows + 7) / 8;             // 8 waves/block
    const int node_blocks  = (NODES + 255) / 256;              // 391
    const int nodeW_blocks = (NODES * 32 + 255) / 256;         // 12500
    const int nh_blocks    = (NODES * HDIM + 255) / 256;       // 25000
    const int edge_blocks  = (NEDGE + 255) / 256;              // 6250
    const int eagg_blocks  = (NEDGE * 16 + 255) / 256;         // 100000

    // -------- Layer 1: h1 = x @ W1 ; GAT aggregate over edge_index1 --------
    gat_gemm_wmma_f32<4><<<gemm_blocks, blk, 0, stream>>>(x, W1, hbuf, nullptr,
                                                          NODES, FDIM);
    gat_scores<<<nodeW_blocks, blk, 0, stream>>>(hbuf, att_src1, att_dst1, asrc, adst);
    gat_init_nodes<<<node_blocks, blk, 0, stream>>>(mbuf, sbuf);
    gat_init_agg<<<nh_blocks, blk, 0, stream>>>(aggbuf, b1);
    gat_edge_max<<<edge_blocks, blk, 0, stream>>>(ei1, ei1 + NEDGE, asrc, adst, ebuf, mbuf);
    gat_edge_exp<<<edge_blocks, blk, 0, stream>>>(ei1 + NEDGE, ebuf, mbuf, sbuf);
    gat_edge_agg<<<eagg_blocks, blk, 0, stream>>>(ei1, ei1 + NEDGE, ebuf, sbuf, hbuf, aggbuf);
    gat_relu<<<nh_blocks, blk, 0, stream>>>(aggbuf, NODES * HDIM);

    // -------- Layer 2: h2 = relu(agg1) @ W2 ; aggregate over edge_index2 ---
    gat_gemm_wmma_f32<4><<<gemm_blocks, blk, 0, stream>>>(aggbuf, W2, hbuf, nullptr,
                                                          NODES, HDIM);
    gat_scores<<<nodeW_blocks, blk, 0, stream>>>(hbuf, att_src2, att_dst2, asrc, adst);
    gat_init_nodes<<<node_blocks, blk, 0, stream>>>(mbuf, sbuf);
    gat_init_agg<<<nh_blocks, blk, 0, stream>>>(aggbuf, b2);
    gat_edge_max<<<edge_blocks, blk, 0, stream>>>(ei2, ei2 + NEDGE, asrc, adst, ebuf, mbuf);
    gat_edge_exp<<<edge_blocks, blk, 0, stream>>>(ei2 + NEDGE, ebuf, mbuf, sbuf);
    gat_edge_agg<<<eagg_blocks, blk, 0, stream>>>(ei2, ei2 + NEDGE, ebuf, sbuf, hbuf, aggbuf);

    // -------- Final linear + log_softmax --------
    gat_gemm_wmma_f32<2><<<gemm_blocks, blk, 0, stream>>>(aggbuf, Wlin, logits, blin,
                                                          NODES, HDIM);
    gat_logsoftmax<<<nodeW_blocks, blk, 0, stream>>>(logits, out);
}